// PerPixelConv_62045097558442
// MI455X (gfx1250) — compile-verified
//
#include <hip/hip_runtime.h>
#include <cstdint>

typedef unsigned int __attribute__((ext_vector_type(4))) v4u;
typedef int          __attribute__((ext_vector_type(8))) v8i;
typedef int          __attribute__((ext_vector_type(4))) v4i;
typedef float        __attribute__((ext_vector_type(4))) f4;

// Per-pixel 3x3 conv:
//   out[b,h,w] = sum_{kw,kh} img_pad[b, h+kh, w+kw] * ker[b, kw*3+kh, h, w]
// One block handles one (b,h) output row: 128 threads x 4 px = 512 = W.
// Image rows h-1..h+1 staged into LDS by the Tensor Data Mover (TDM).
__global__ __launch_bounds__(128) void ppconv_kernel(
    const float* __restrict__ ker,
    const float* __restrict__ img,
    float* __restrict__ out)
{
    constexpr int Wd = 512;
    constexpr int Hd = 512;
    __shared__ float tile[3 * Wd];   // 3 image rows, packed (row stride = 512 floats)

    const int tid = threadIdx.x;          // 0..127
    const int row = blockIdx.x;           // 0 .. 16*512-1
    const int b   = row >> 9;
    const int h   = row & (Hd - 1);

    // ---- zero-fill LDS halo rows that fall outside the image ----
    if (h == 0) {
        #pragma unroll
        for (int j = 0; j < 4; ++j) tile[0 * Wd + tid * 4 + j] = 0.0f;
    }
    if (h == Hd - 1) {
        #pragma unroll
        for (int j = 0; j < 4; ++j) tile[2 * Wd + tid * 4 + j] = 0.0f;
    }

    // ---- TDM async tensor load of the valid image rows into LDS ----
    const int r0     = h - 1;                       // first tile row (image coords)
    const int rstart = (r0 < 0) ? 0 : r0;
    const int rend   = (h + 1 > Hd - 1) ? (Hd - 1) : (h + 1);
    const int nrows  = rend - rstart + 1;           // 2 at borders, else 3
    const int ldsRow0 = rstart - r0;                // 0 or 1

    if (tid < 32) {  // one wave issues the DMA
        // 'tile' is the only LDS object -> group-segment offset 0.
        const uint32_t lds_off =
            (uint32_t)((uint32_t)ldsRow0 * (uint32_t)Wd * sizeof(float));
        const uint64_t gaddr =
            (uint64_t)(uintptr_t)(img + ((size_t)b * Hd + (size_t)rstart) * Wd);

        // D# group 0 (ISA 8.3): count=1 | lds_addr | global_addr[56:0] | type=2
        v4u g0;
        g0.x = 1u;                                           // count=1, user mode
        g0.y = lds_off;                                      // lds_addr (bytes)
        g0.z = (uint32_t)(gaddr & 0xFFFFFFFFu);              // global_addr[31:0]
        g0.w = (uint32_t)((gaddr >> 32) & 0x01FFFFFFu)       // global_addr[56:32]
             | (2u << 30);                                   // type = 2 ("image")

        // D# group 1 (ISA 8.4): data_size=4B, 2-D tile 512 x nrows
        v8i g1;
        g1[0] = (int)(2u << 16);                 // wg_mask=0, data_size=2 (4B)
        g1[1] = (int)(512u << 16);               // tensor_dim0 = 512 (bits 79:48 lo)
        g1[2] = (int)(((uint32_t)nrows) << 16);  // tensor_dim0 hi=0, tensor_dim1=nrows
        g1[3] = (int)(512u << 16);               // tensor_dim1 hi=0, tile_dim0=512
        g1[4] = nrows;                           // tile_dim1=nrows, tile_dim2=0
        g1[5] = 512;                             // tensor_dim0_stride = 512 (lo 32)
        g1[6] = 0;                               // stride hi, dim1_stride lo (unused 2D)
        g1[7] = 0;

        const v4i z4 = (v4i)0;                   // groups 2/3 unused for 2-D tensor
        const v8i z8 = (v8i)0;                   // extra group (6-arg toolchain form)
        __builtin_amdgcn_tensor_load_to_lds(g0, g1, z4, z4, z8, /*cpol=*/0);
        __builtin_amdgcn_s_wait_tensorcnt(0);
    }
    __syncthreads();

    // ---- compute 4 output pixels per thread ----
    const int w0 = tid * 4;

    // 18 image taps: rows 0..2 (image rows h-1..h+1), cols w0-1 .. w0+4
    float tv[3][6];
    #pragma unroll
    for (int r = 0; r < 3; ++r) {
        #pragma unroll
        for (int c = 0; c < 6; ++c) {
            const int gc = w0 - 1 + c;
            tv[r][c] = (gc >= 0 && gc < Wd) ? tile[r * Wd + gc] : 0.0f;
        }
    }

    const size_t plane = (size_t)Wd * Hd;
    const float* kbase = ker + (size_t)b * 9 * plane + (size_t)h * Wd + w0;

    f4 acc = (f4)0.0f;
    #pragma unroll
    for (int kw = 0; kw < 3; ++kw) {
        #pragma unroll
        for (int kh = 0; kh < 3; ++kh) {
            const int idx = kw * 3 + kh;
            // single-use weight stream: non-temporal 128-bit load
            const f4 kv = __builtin_nontemporal_load(
                reinterpret_cast<const f4*>(kbase + (size_t)idx * plane));
            #pragma unroll
            for (int j = 0; j < 4; ++j)
                acc[j] = fmaf(kv[j], tv[kh][j + kw], acc[j]);
        }
    }

    f4* op = reinterpret_cast<f4*>(out + (size_t)b * plane + (size_t)h * Wd + w0);
    __builtin_nontemporal_store(acc, op);
}

extern "C" void kernel_launch(void* const* d_in, const int* in_sizes, int n_in,
                              void* d_out, int out_size, void* d_ws, size_t ws_size,
                              hipStream_t stream) {
    const float* ker = (const float*)d_in[0];   // [16, 9, 512, 512] fp32
    const float* img = (const float*)d_in[1];   // [16, 1, 512, 512] fp32
    float* out = (float*)d_out;                 // [16, 1, 512, 512] fp32

    const int B = 16, H = 512;
    dim3 grid(B * H);      // one block per (b, h) row
    dim3 block(128);       // 4 waves; 128 x 4 px = 512 = W
    ppconv_kernel<<<grid, block, 0, stream>>>(ker, img, out);
}